// Detect_Flip_73504070303849
// MI455X (gfx1250) — compile-verified
//
#include <hip/hip_runtime.h>
#include <hip/hip_bf16.h>

#define BATCH   32
#define NANCH   8732
#define NCLS    21
#define N2      (2 * NANCH)      // 17464 anchors after TTA concat (divisible by 4)
#define TOPK    200
#define CONF_T  0.01f
#define NMS_T   0.45f
#define TPB     256
#define NWAVES  (TPB / 32)

// ---- CDNA5 async global->LDS path (guarded; fallback = plain loads) -------
#if __has_builtin(__builtin_amdgcn_global_load_async_to_lds_b32) && \
    __has_builtin(__builtin_amdgcn_s_wait_asynccnt)
#define HAVE_ASYNC 1
#else
#define HAVE_ASYNC 0
#endif

#if __has_builtin(__builtin_amdgcn_global_load_async_to_lds_b128) && \
    __has_builtin(__builtin_amdgcn_s_wait_asynccnt)
#define HAVE_ASYNC128 1
#else
#define HAVE_ASYNC128 0
#endif

#if HAVE_ASYNC
typedef __attribute__((address_space(1))) int g_as_int;   // global
typedef __attribute__((address_space(3))) int l_as_int;   // LDS

__device__ __forceinline__ void async_load_f32_to_lds(const float* g, float* l) {
  __builtin_amdgcn_global_load_async_to_lds_b32(
      (g_as_int*)g, (l_as_int*)l, /*offset=*/0, /*cpol=*/0);
}
#endif

#if HAVE_ASYNC128
typedef int int4v __attribute__((ext_vector_type(4)));
typedef __attribute__((address_space(1))) int4v g_as_int4;  // global
typedef __attribute__((address_space(3))) int4v l_as_int4;  // LDS

__device__ __forceinline__ void async_load_b128_to_lds(const float* g, float* l) {
  __builtin_amdgcn_global_load_async_to_lds_b128(
      (g_as_int4*)g, (l_as_int4*)l, /*offset=*/0, /*cpol=*/0);
}
#endif

#if HAVE_ASYNC || HAVE_ASYNC128
__device__ __forceinline__ void wait_async0() {
  __builtin_amdgcn_s_wait_asynccnt(0);
}
#endif

// ---------------------------------------------------------------------------
// Kernel 1: softmax (both views) -> scores_t [B][C][2N] (class-major, coalesced)
//           decode + flip        -> boxes    [B][2N][4]
// ---------------------------------------------------------------------------
__global__ void __launch_bounds__(TPB)
ssd_preprocess(const float* __restrict__ loc1, const float* __restrict__ conf1,
               const float* __restrict__ loc2, const float* __restrict__ conf2,
               const float* __restrict__ dbox,
               float* __restrict__ scores_t, float* __restrict__ boxes) {
  int gid = blockIdx.x * TPB + threadIdx.x;
  if (gid >= BATCH * NANCH) return;
  int b = gid / NANCH;
  int n = gid - b * NANCH;

  const float* c1 = conf1 + (size_t)(b * NANCH + n) * NCLS;
  const float* c2 = conf2 + (size_t)(b * NANCH + n) * NCLS;
  __builtin_prefetch(c1 + NCLS, 0, 1);   // global_prefetch_b8
  __builtin_prefetch(c2 + NCLS, 0, 1);

  size_t sbase = (size_t)b * NCLS * N2 + n;   // element (b, c, n) at sbase + c*N2

  // --- softmax view 1 ---
  {
    float l[NCLS];
    float m = -1e30f;
#pragma unroll
    for (int c = 0; c < NCLS; ++c) { l[c] = c1[c]; m = fmaxf(m, l[c]); }
    float s = 0.f;
#pragma unroll
    for (int c = 0; c < NCLS; ++c) { l[c] = expf(l[c] - m); s += l[c]; }
    float inv = 1.0f / s;
#pragma unroll
    for (int c = 0; c < NCLS; ++c) scores_t[sbase + (size_t)c * N2] = l[c] * inv;
  }
  // --- softmax view 2 (stored at anchor offset +NANCH) ---
  {
    float l[NCLS];
    float m = -1e30f;
#pragma unroll
    for (int c = 0; c < NCLS; ++c) { l[c] = c2[c]; m = fmaxf(m, l[c]); }
    float s = 0.f;
#pragma unroll
    for (int c = 0; c < NCLS; ++c) { l[c] = expf(l[c] - m); s += l[c]; }
    float inv = 1.0f / s;
#pragma unroll
    for (int c = 0; c < NCLS; ++c)
      scores_t[sbase + (size_t)c * N2 + NANCH] = l[c] * inv;
  }

  // --- decode boxes (SSD variances 0.1 / 0.2) ---
  float d0 = dbox[n * 4 + 0], d1 = dbox[n * 4 + 1];
  float d2 = dbox[n * 4 + 2], d3 = dbox[n * 4 + 3];

  {  // view 1: plain
    const float* L = loc1 + (size_t)(b * NANCH + n) * 4;
    float cx = d0 + L[0] * 0.1f * d2;
    float cy = d1 + L[1] * 0.1f * d3;
    float w  = d2 * expf(L[2] * 0.2f);
    float h  = d3 * expf(L[3] * 0.2f);
    float* o = boxes + ((size_t)b * N2 + n) * 4;
    o[0] = cx - 0.5f * w; o[1] = cy - 0.5f * h;
    o[2] = cx + 0.5f * w; o[3] = cy + 0.5f * h;
  }
  {  // view 2: decode then horizontal flip (x1'=1-x2, x2'=1-x1)
    const float* L = loc2 + (size_t)(b * NANCH + n) * 4;
    float cx = d0 + L[0] * 0.1f * d2;
    float cy = d1 + L[1] * 0.1f * d3;
    float w  = d2 * expf(L[2] * 0.2f);
    float h  = d3 * expf(L[3] * 0.2f);
    float* o = boxes + ((size_t)b * N2 + NANCH + n) * 4;
    o[0] = 1.0f - (cx + 0.5f * w); o[1] = cy - 0.5f * h;
    o[2] = 1.0f - (cx - 0.5f * w); o[3] = cy + 0.5f * h;
  }
}

// ---------------------------------------------------------------------------
// Kernel 2: zero the output (harness poisons d_out; ref output is mostly 0)
// ---------------------------------------------------------------------------
__global__ void ssd_zero(float* __restrict__ out, int n) {
  for (int i = blockIdx.x * blockDim.x + threadIdx.x; i < n; i += gridDim.x * blockDim.x)
    out[i] = 0.0f;
}

// ---------------------------------------------------------------------------
// Kernel 3: per-(batch,class) top-200 + sequential NMS + compacted scatter.
// One 256-thread (8-wave) block per (b, cls), cls in [1, NCLS).
// Entire score row lives in LDS (CDNA5: 320 KB/WGP).
// ---------------------------------------------------------------------------
__global__ void __launch_bounds__(TPB)
ssd_topk_nms(const float* __restrict__ scores_t,
             const float* __restrict__ boxes,
             float* __restrict__ out) {
  __shared__ __align__(16) float s_sc[N2];   // masked scores (68.2 KB)
  __shared__ float s_tv[TPB];         // per-thread running max value
  __shared__ int   s_ti[TPB];         // per-thread running max index
  __shared__ float s_rv[NWAVES];
  __shared__ int   s_ri[NWAVES];
  __shared__ int   s_pick;
  __shared__ float s_cs[TOPK];        // candidate scores (desc)
  __shared__ int   s_ci[TOPK];        // candidate anchor indices
  __shared__ float s_bx[TOPK][4];     // candidate boxes
  __shared__ float s_ar[TOPK];        // candidate areas
  __shared__ int   s_sup[TOPK];       // suppressed flags

  const int tid  = threadIdx.x;
  const int lane = tid & 31;
  const int wid  = tid >> 5;
  const int bc   = blockIdx.x;                 // 0 .. B*(C-1)-1
  const int b    = bc / (NCLS - 1);
  const int cls  = 1 + (bc - b * (NCLS - 1));

  const float* gsc = scores_t + ((size_t)b * NCLS + cls) * N2;
  const float* gbx = boxes + (size_t)b * N2 * 4;

  // ---- stage score row into LDS (async b128 > async b32 > plain) ----
#if HAVE_ASYNC128
  for (int i4 = tid; i4 < N2 / 4; i4 += TPB)
    async_load_b128_to_lds(gsc + i4 * 4, &s_sc[i4 * 4]);
  wait_async0();
  __syncthreads();
#elif HAVE_ASYNC
  for (int i = tid; i < N2; i += TPB)
    async_load_f32_to_lds(gsc + i, &s_sc[i]);
  wait_async0();
  __syncthreads();
#else
  for (int i = tid; i < N2; i += TPB)
    s_sc[i] = gsc[i];
  __syncthreads();
#endif

  // ---- mask in place: score if > thresh else -1 (vectorized ds_b128) ----
  {
    float4* s4 = (float4*)s_sc;
    for (int i4 = tid; i4 < N2 / 4; i4 += TPB) {
      float4 v = s4[i4];
      v.x = (v.x > CONF_T) ? v.x : -1.0f;
      v.y = (v.y > CONF_T) ? v.y : -1.0f;
      v.z = (v.z > CONF_T) ? v.z : -1.0f;
      v.w = (v.w > CONF_T) ? v.w : -1.0f;
      s4[i4] = v;
    }
  }
  __syncthreads();

  // ---- per-thread partial max over its strided subset ----
  {
    float bv = -1e30f; int bi = 0x7fffffff;
    for (int i = tid; i < N2; i += TPB) {
      float v = s_sc[i];
      if (v > bv) { bv = v; bi = i; }   // strict '>' keeps lowest index on ties
    }
    s_tv[tid] = bv; s_ti[tid] = bi;
  }
  __syncthreads();

  // ---- incremental iterative arg-max: 200 picks ----
  for (int k = 0; k < TOPK; ++k) {
    float v = s_tv[tid];
    int   ix = s_ti[tid];
#pragma unroll
    for (int off = 16; off > 0; off >>= 1) {   // wave32 reduce
      float ov = __shfl_down(v, off);
      int   oi = __shfl_down(ix, off);
      if (ov > v || (ov == v && oi < ix)) { v = ov; ix = oi; }
    }
    if (lane == 0) { s_rv[wid] = v; s_ri[wid] = ix; }
    __syncthreads();
    if (tid == 0) {
      float wv = s_rv[0]; int wi = s_ri[0];
#pragma unroll
      for (int w = 1; w < NWAVES; ++w) {
        float xv = s_rv[w]; int xi = s_ri[w];
        if (xv > wv || (xv == wv && xi < wi)) { wv = xv; wi = xi; }
      }
      s_cs[k] = wv;
      s_ci[k] = wi;
      s_sc[wi] = -2.0f;                // below mask value: never re-picked
      s_pick = wi;
    }
    __syncthreads();
    int wi = s_pick;
    if (tid == (wi & (TPB - 1))) {     // only the owner's subset changed
      float nbv = -1e30f; int nbi = 0x7fffffff;
      for (int i = tid; i < N2; i += TPB) {
        float vv = s_sc[i];
        if (vv > nbv) { nbv = vv; nbi = i; }
      }
      s_tv[tid] = nbv; s_ti[tid] = nbi;
    }
    __syncthreads();
  }

  // ---- fetch candidate boxes, areas; clear suppression ----
  if (tid < TOPK) {
    const float* bp = gbx + (size_t)s_ci[tid] * 4;
    float x1 = bp[0], y1 = bp[1], x2 = bp[2], y2 = bp[3];
    s_bx[tid][0] = x1; s_bx[tid][1] = y1; s_bx[tid][2] = x2; s_bx[tid][3] = y2;
    s_ar[tid] = (x2 - x1) * (y2 - y1);
    s_sup[tid] = 0;
  }
  __syncthreads();

  // ---- sequential NMS scan; IoU row computed in parallel per step ----
  int cnt = 0;
  const size_t obase = ((size_t)b * NCLS + cls) * (size_t)TOPK * 5;
  for (int j = 0; j < TOPK; ++j) {
    float sj   = s_cs[j];                         // uniform LDS broadcast
    int   supj = s_sup[j];
    int   keepj = (sj > CONF_T) && !supj;
    __syncthreads();                              // reads before writes below
    if (keepj) {
      if (tid < TOPK) {
        float xx1 = fmaxf(s_bx[j][0], s_bx[tid][0]);
        float yy1 = fmaxf(s_bx[j][1], s_bx[tid][1]);
        float xx2 = fminf(s_bx[j][2], s_bx[tid][2]);
        float yy2 = fminf(s_bx[j][3], s_bx[tid][3]);
        float inter = fmaxf(xx2 - xx1, 0.0f) * fmaxf(yy2 - yy1, 0.0f);
        float uni = s_ar[j] + s_ar[tid] - inter;
        if (inter / uni > NMS_T) s_sup[tid] = 1;
      }
      if (tid < 5) {
        float val = (tid == 0) ? sj : s_bx[j][tid - 1];
        out[obase + (size_t)cnt * 5 + tid] = val;
      }
      ++cnt;                                      // uniform across block
    }
    __syncthreads();
  }
}

// ---------------------------------------------------------------------------
extern "C" void kernel_launch(void* const* d_in, const int* in_sizes, int n_in,
                              void* d_out, int out_size, void* d_ws, size_t ws_size,
                              hipStream_t stream) {
  const float* loc1  = (const float*)d_in[0];
  const float* conf1 = (const float*)d_in[1];
  const float* loc2  = (const float*)d_in[2];
  const float* conf2 = (const float*)d_in[3];
  const float* dbox  = (const float*)d_in[4];
  float* out = (float*)d_out;

  // workspace: scores_t [B][C][2N] then boxes [B][2N][4]  (~56 MB total)
  float* scores_t = (float*)d_ws;
  float* boxes    = scores_t + (size_t)BATCH * NCLS * N2;

  const int total = BATCH * NANCH;
  ssd_preprocess<<<(total + TPB - 1) / TPB, TPB, 0, stream>>>(
      loc1, conf1, loc2, conf2, dbox, scores_t, boxes);

  ssd_zero<<<(out_size + TPB - 1) / TPB, TPB, 0, stream>>>(out, out_size);

  ssd_topk_nms<<<BATCH * (NCLS - 1), TPB, 0, stream>>>(scores_t, boxes, out);
}